// AEAttention_88038239634067
// MI455X (gfx1250) — compile-verified
//
#include <hip/hip_runtime.h>
#include <math.h>

typedef float v2f __attribute__((ext_vector_type(2)));
typedef float v8f __attribute__((ext_vector_type(8)));

__device__ __forceinline__ float gelu_exact(float x) {
    // torch nn.GELU default: 0.5*x*(1+erf(x/sqrt(2)))
    return 0.5f * x * (1.0f + erff(x * 0.70710678118654752440f));
}

// ---------------------------------------------------------------------------
// One 16x16 f32 tile via V_WMMA_F32_16X16X4_F32, A staged in LDS, B in global.
// A-frag (ISA 7.12.2, 32-bit A 16x4): lanes 0-15 K={k0,k0+1}, lanes 16-31
// K={k0+2,k0+3}, row = lane&15. B mirrors (rows striped across lanes).
// C/D v8f: row = v + 8*(lane>>4), col = lane&15.
// Incremental pointers keep per-iteration address math to two adds.
// ---------------------------------------------------------------------------
__device__ __forceinline__ v8f wmma_tile(const float* __restrict__ Al, int lda,
                                         const float* __restrict__ Bg, int ldb,
                                         int col0, int K, int lane)
{
    const int half = lane >> 4;
    const int m    = lane & 15;
    const float* ap = Al + m * lda + 2 * half;
    const float* bp = Bg + (2 * half) * ldb + col0 + m;
    v8f c = {};
    for (int k0 = 0; k0 < K; k0 += 4) {
        v2f a = *(const v2f*)ap;           // 8B-aligned (lda even)
        v2f b;
        b.x = bp[0];
        b.y = bp[ldb];
        c = __builtin_amdgcn_wmma_f32_16x16x4_f32(
                /*neg_a=*/false, a, /*neg_b=*/false, b,
                /*c_mod=*/(short)0, c, /*reuse_a=*/false, /*reuse_b=*/false);
        ap += 4;
        bp += 4 * ldb;
    }
    return c;
}

// ---------------------------------------------------------------------------
// K1: h = gelu(x@U_w + U_b); split; LayerNorm(z) fused in-block.
// Grid 32 x 256 threads (8 waves). Each block: 16 rows, all 768 cols.
// ---------------------------------------------------------------------------
__global__ __launch_bounds__(256)
void fused_up_ln(const float* __restrict__ x, const float* __restrict__ U_w,
                 const float* __restrict__ U_b, const float* __restrict__ ln_w,
                 const float* __restrict__ ln_b, float* __restrict__ xh,
                 float* __restrict__ zln)
{
    __shared__ float xs[16 * 384];   // A tile
    __shared__ float zs[16 * 384];   // z half of output tile
    __shared__ float red[256];
    __shared__ float mrow[16], rrow[16];

    const int t    = threadIdx.x;
    const int wave = t >> 5, lane = t & 31;
    const int half = lane >> 4, m = lane & 15;
    const int r0   = blockIdx.x * 16;

    for (int idx = t; idx < 16 * 384; idx += 256) xs[idx] = x[r0 * 384 + idx];
    __syncthreads();

    for (int it = 0; it < 6; ++it) {                 // 48 tiles / 8 waves
        const int col0 = (wave * 6 + it) * 16;
        v8f c = wmma_tile(xs, 384, U_w, 768, col0, 384, lane);
        const float bv = U_b[col0 + m];
        if (col0 < 384) {
#pragma unroll
            for (int v = 0; v < 8; ++v)
                xh[(r0 + v + 8 * half) * 384 + col0 + m] = gelu_exact(c[v] + bv);
        } else {
#pragma unroll
            for (int v = 0; v < 8; ++v)
                zs[(v + 8 * half) * 384 + (col0 - 384) + m] = gelu_exact(c[v] + bv);
        }
    }
    __syncthreads();

    // LayerNorm: 16 threads per row, deterministic tree reductions.
    const int r = t >> 4, c0 = t & 15;
    float s = 0.0f;
    for (int k = 0; k < 24; ++k) s += zs[r * 384 + c0 + 16 * k];
    red[t] = s; __syncthreads();
    for (int o = 8; o > 0; o >>= 1) { if (c0 < o) red[t] += red[t + o]; __syncthreads(); }
    if (c0 == 0) mrow[r] = red[t] * (1.0f / 384.0f);
    __syncthreads();
    const float mu = mrow[r];
    float s2 = 0.0f;
    for (int k = 0; k < 24; ++k) { const float d = zs[r * 384 + c0 + 16 * k] - mu; s2 += d * d; }
    red[t] = s2; __syncthreads();
    for (int o = 8; o > 0; o >>= 1) { if (c0 < o) red[t] += red[t + o]; __syncthreads(); }
    if (c0 == 0) rrow[r] = rsqrtf(red[t] * (1.0f / 384.0f) + 1e-5f);
    __syncthreads();
    const float rstd = rrow[r];
    for (int k = 0; k < 24; ++k) {
        const int c = c0 + 16 * k;
        zln[(r0 + r) * 384 + c] = (zs[r * 384 + c] - mu) * rstd * ln_w[c] + ln_b[c];
    }
}

// ---------------------------------------------------------------------------
// K2: per 16 rows: enc GEMM(+gelu) -> dec GEMM -> ddiag=<preds,zln>,
// d0=<p0,zln> with p0 = gelu(enc_b)@dec_w + dec_b recomputed in-block.
// ---------------------------------------------------------------------------
__global__ __launch_bounds__(256)
void fused_ae(const float* __restrict__ zln, const float* __restrict__ enc_w,
              const float* __restrict__ enc_b, const float* __restrict__ dec_w,
              const float* __restrict__ dec_b, float* __restrict__ d0,
              float* __restrict__ ddiag)
{
    __shared__ float zl[16 * 384];   // 24.0 KB
    __shared__ float hm[16 * 192];   // 12.0 KB (reused as reduction scratch)
    __shared__ float pr[16 * 384];   // 24.0 KB
    __shared__ float ge[192];
    __shared__ float p0s[384];

    const int t    = threadIdx.x;
    const int wave = t >> 5, lane = t & 31;
    const int half = lane >> 4, m = lane & 15;
    const int r0   = blockIdx.x * 16;

    for (int idx = t; idx < 16 * 384; idx += 256) zl[idx] = zln[r0 * 384 + idx];
    if (t < 192) ge[t] = gelu_exact(enc_b[t]);
    __syncthreads();

    // p0 (384 values across 256 threads)
    {
        float a0 = dec_b[t];
        float a1 = (t < 128) ? dec_b[t + 256] : 0.0f;
        for (int h = 0; h < 192; ++h) {
            const float g = ge[h];
            a0 += g * dec_w[h * 384 + t];
            if (t < 128) a1 += g * dec_w[h * 384 + t + 256];
        }
        p0s[t] = a0;
        if (t < 128) p0s[t + 256] = a1;
    }

    // enc: hm = gelu(zl @ enc_w + enc_b)   (16x384)(384x192)
    for (int tc = wave; tc < 12; tc += 8) {
        const int col0 = tc * 16;
        v8f c = wmma_tile(zl, 384, enc_w, 192, col0, 384, lane);
        const float bv = enc_b[col0 + m];
#pragma unroll
        for (int v = 0; v < 8; ++v)
            hm[(v + 8 * half) * 192 + col0 + m] = gelu_exact(c[v] + bv);
    }
    __syncthreads();

    // dec: pr = hm @ dec_w + dec_b         (16x192)(192x384)
    for (int tc = wave; tc < 24; tc += 8) {
        const int col0 = tc * 16;
        v8f c = wmma_tile(hm, 192, dec_w, 384, col0, 192, lane);
        const float bv = dec_b[col0 + m];
#pragma unroll
        for (int v = 0; v < 8; ++v)
            pr[(v + 8 * half) * 384 + col0 + m] = c[v] + bv;
    }
    __syncthreads();

    // hm is dead now -> alias reduction scratch onto it (stay under 64KB LDS)
    float* red  = hm;
    float* red2 = hm + 256;
    const int r = t >> 4, c0 = t & 15;
    float sdd = 0.0f, sd0 = 0.0f;
    for (int k = 0; k < 24; ++k) {
        const int c  = c0 + 16 * k;
        const float zv = zl[r * 384 + c];
        sdd += pr[r * 384 + c] * zv;
        sd0 += p0s[c] * zv;
    }
    red[t] = sdd; red2[t] = sd0; __syncthreads();
    for (int o = 8; o > 0; o >>= 1) {
        if (c0 < o) { red[t] += red[t + o]; red2[t] += red2[t + o]; }
        __syncthreads();
    }
    if (c0 == 0) { ddiag[r0 + r] = red[t]; d0[r0 + r] = red2[t]; }
}

// ---------------------------------------------------------------------------
// K3: per 16 rows (single batch b): softmax rows from d0/ddiag -> att (LDS)
// -> WMMA att@xh_b -> oa (LDS) -> WMMA oa@V_w + V_b -> d_out.
// ---------------------------------------------------------------------------
__global__ __launch_bounds__(256)
void fused_attn_out(const float* __restrict__ d0, const float* __restrict__ ddiag,
                    const float* __restrict__ xh, const float* __restrict__ V_w,
                    const float* __restrict__ V_b, float* __restrict__ out)
{
    __shared__ float d0b[128];
    __shared__ float dd[16];
    __shared__ float att[16 * 128];  // 8 KB
    __shared__ float oa[16 * 384];   // 24 KB
    __shared__ float red[256];
    __shared__ float mx[16], sm[16];

    const int t    = threadIdx.x;
    const int wave = t >> 5, lane = t & 31;
    const int half = lane >> 4, m = lane & 15;
    const int r0   = blockIdx.x * 16;
    const int b    = r0 >> 7;        // 128 rows per batch, 16 | 128
    const int i0   = r0 & 127;

    if (t < 128) d0b[t] = d0[b * 128 + t];
    if (t < 16)  dd[t]  = ddiag[r0 + t];
    __syncthreads();

    // softmax over j of dist[i,j] = (j==i) ? ddiag[i] : d0[j]
    const int r = t >> 4, c0 = t & 15;
    const int irow = i0 + r;
    float mmax = -3.402823466e38f;
    for (int k = 0; k < 8; ++k) {
        const int j = c0 + 16 * k;
        const float val = (j == irow) ? dd[r] : d0b[j];
        mmax = fmaxf(mmax, val);
    }
    red[t] = mmax; __syncthreads();
    for (int o = 8; o > 0; o >>= 1) { if (c0 < o) red[t] = fmaxf(red[t], red[t + o]); __syncthreads(); }
    if (c0 == 0) mx[r] = red[t];
    __syncthreads();
    const float mrow = mx[r];
    float ssum = 0.0f;
    for (int k = 0; k < 8; ++k) {
        const int j = c0 + 16 * k;
        const float val = (j == irow) ? dd[r] : d0b[j];
        ssum += expf(val - mrow);
    }
    red[t] = ssum; __syncthreads();
    for (int o = 8; o > 0; o >>= 1) { if (c0 < o) red[t] += red[t + o]; __syncthreads(); }
    if (c0 == 0) sm[r] = red[t];
    __syncthreads();
    const float inv = 1.0f / sm[r];
    for (int k = 0; k < 8; ++k) {
        const int j = c0 + 16 * k;
        const float val = (j == irow) ? dd[r] : d0b[j];
        att[r * 128 + j] = expf(val - mrow) * inv;
    }
    __syncthreads();

    // oa = att @ xh_b   (16x128)(128x384)
    const float* xb = xh + b * (128 * 384);
    for (int tc = wave; tc < 24; tc += 8) {
        const int col0 = tc * 16;
        v8f c = wmma_tile(att, 128, xb, 384, col0, 128, lane);
#pragma unroll
        for (int v = 0; v < 8; ++v)
            oa[(v + 8 * half) * 384 + col0 + m] = c[v];
    }
    __syncthreads();

    // out = oa @ V_w + V_b   (16x384)(384x384)
    for (int tc = wave; tc < 24; tc += 8) {
        const int col0 = tc * 16;
        v8f c = wmma_tile(oa, 384, V_w, 384, col0, 384, lane);
        const float bv = V_b[col0 + m];
#pragma unroll
        for (int v = 0; v < 8; ++v)
            out[(r0 + v + 8 * half) * 384 + col0 + m] = c[v] + bv;
    }
}

// ---------------------------------------------------------------------------
extern "C" void kernel_launch(void* const* d_in, const int* in_sizes, int n_in,
                              void* d_out, int out_size, void* d_ws, size_t ws_size,
                              hipStream_t stream)
{
    const float* x     = (const float*)d_in[0];   // (4,128,384)
    const float* U_w   = (const float*)d_in[1];   // (384,768)
    const float* U_b   = (const float*)d_in[2];   // (768)
    const float* ln_w  = (const float*)d_in[3];   // (384)
    const float* ln_b  = (const float*)d_in[4];   // (384)
    const float* enc_w = (const float*)d_in[5];   // (384,192)
    const float* enc_b = (const float*)d_in[6];   // (192)
    const float* dec_w = (const float*)d_in[7];   // (192,384)
    const float* dec_b = (const float*)d_in[8];   // (384)
    const float* V_w   = (const float*)d_in[9];   // (384,384)
    const float* V_b   = (const float*)d_in[10];  // (384)
    float* out = (float*)d_out;                   // (4,128,384)

    float* ws   = (float*)d_ws;
    float* xh   = ws;                  // 512*384
    float* zln  = xh  + 512 * 384;     // 512*384
    float* d0v  = zln + 512 * 384;     // 512
    float* ddg  = d0v + 512;           // 512

    fused_up_ln   <<<32, 256, 0, stream>>>(x, U_w, U_b, ln_w, ln_b, xh, zln);
    fused_ae      <<<32, 256, 0, stream>>>(zln, enc_w, enc_b, dec_w, dec_b, d0v, ddg);
    fused_attn_out<<<32, 256, 0, stream>>>(d0v, ddg, xh, V_w, V_b, out);

    (void)in_sizes; (void)n_in; (void)out_size; (void)ws_size;
}